// Downsample_11828339933399
// MI455X (gfx1250) — compile-verified
//
#include <hip/hip_runtime.h>
#include <stdint.h>

// Depthwise 4x4 FIR, stride 2, pad 1:  x(16,256,256,256) f32 -> out(16,256,128,128) f32
// Memory-bound (AI ~1.6 F/B) => single-pass HBM streaming via async global->LDS staging.

#define TH       16              // output rows per block
#define NROWS    (2 * TH + 2)    // 34 staged input rows (incl. halo)
#define RSTRIDE  264             // floats per padded LDS row; data starts at +4 (16B aligned)

#if defined(__HIP_DEVICE_COMPILE__)
  #if __has_builtin(__builtin_amdgcn_global_load_async_to_lds_b128)
    #define USE_ASYNC_LDS 1
  #endif
  #if __has_builtin(__builtin_amdgcn_s_wait_asynccnt)
    #define HAVE_WAIT_ASYNC 1
  #endif
#endif

// Exact types for the async-to-LDS builtin:
//   arg0: v4i32 addrspace(1)*  (prints as "... int __device__ *" in HIP diagnostics)
//   arg1: v4i32 addrspace(3)*
typedef int vint4 __attribute__((vector_size(4 * sizeof(int))));
typedef __attribute__((address_space(1))) vint4 g_vint4;
typedef __attribute__((address_space(3))) vint4 l_vint4;

__global__ __launch_bounds__(256)
void downsample2x_fir4(const float* __restrict__ x,
                       const float* __restrict__ kf,
                       float* __restrict__ out)
{
    __shared__ __align__(16) float smem[NROWS * RSTRIDE];   // ~35 KB

    const int tid   = threadIdx.x;
    const int plane = blockIdx.x >> 3;       // n*C + c   (4096 planes)
    const int ytile = blockIdx.x & 7;        // 8 tiles of TH output rows
    const int y0    = ytile * TH;            // first output row of tile
    const int row0  = 2 * y0 - 1;            // first staged input row (pad=1)

    const float* gplane = x + (size_t)plane * (256 * 256);

    // ---- stage NROWS x 256 floats into padded LDS rows (async DMA path) ----
    for (int c = tid; c < NROWS * 64; c += 256) {
        const int r  = c >> 6;               // staged row 0..33
        const int c4 = c & 63;               // 16-byte chunk within row
        const int g  = row0 + r;             // global input row
        float* dst = &smem[r * RSTRIDE + 4 + c4 * 4];
        if (g >= 0 && g < 256) {
            const float* src = gplane + (size_t)g * 256 + c4 * 4;
#if USE_ASYNC_LDS
            __builtin_amdgcn_global_load_async_to_lds_b128(
                (g_vint4*)(uintptr_t)src,
                (l_vint4*)(uintptr_t)dst,
                0 /*imm offset*/, 0 /*cpol*/);
#else
            *(float4*)dst = *(const float4*)src;
#endif
        } else {
            *(float4*)dst = make_float4(0.f, 0.f, 0.f, 0.f);   // zero padding rows
        }
        if (c4 == 0)  smem[r * RSTRIDE + 3]   = 0.f;   // left pad col (-1)
        if (c4 == 63) smem[r * RSTRIDE + 260] = 0.f;   // right pad col (256)
    }

#if USE_ASYNC_LDS
  #if HAVE_WAIT_ASYNC
    __builtin_amdgcn_s_wait_asynccnt(0);
  #else
    asm volatile("s_wait_asynccnt 0x0" ::: "memory");
  #endif
#endif
    __syncthreads();

    // ---- 4x4 weights: uniform address -> scalar (SGPR) loads ----
    float w[16];
#pragma unroll
    for (int i = 0; i < 16; ++i) w[i] = kf[i];

    // ---- compute TH x 128 outputs; each thread does TH/2 rows at one column ----
    const int tx = tid & 127;                // output column
    const int tg = tid >> 7;                 // row phase 0/1
    float* oplane = out + ((size_t)plane * 128 + y0) * 128;

#pragma unroll
    for (int k = 0; k < TH / 2; ++k) {
        const int yl = tg + 2 * k;           // local output row 0..15
        float acc = 0.f;
#pragma unroll
        for (int j = 0; j < 4; ++j) {
            // output row y uses input rows 2y-1..2y+2  ->  LDS rows 2*yl+j
            // output col x uses input cols 2x-1..2x+2  ->  padded index 3+2x ..
            const float* rp = &smem[(2 * yl + j) * RSTRIDE + 3 + 2 * tx];
            acc = fmaf(w[4 * j + 0], rp[0], acc);
            acc = fmaf(w[4 * j + 1], rp[1], acc);
            acc = fmaf(w[4 * j + 2], rp[2], acc);
            acc = fmaf(w[4 * j + 3], rp[3], acc);
        }
        oplane[yl * 128 + tx] = acc;
    }
}

extern "C" void kernel_launch(void* const* d_in, const int* in_sizes, int n_in,
                              void* d_out, int out_size, void* d_ws, size_t ws_size,
                              hipStream_t stream) {
    const float* x  = (const float*)d_in[0];   // (16,256,256,256) f32
    const float* kf = (const float*)d_in[1];   // (4,4) f32 FIR kernel
    float* out = (float*)d_out;                // (16,256,128,128) f32

    dim3 grid(16 * 256 * 8, 1, 1);             // planes * y-tiles = 32768 blocks
    dim3 block(256, 1, 1);                     // 8 waves (wave32)
    downsample2x_fir4<<<grid, block, 0, stream>>>(x, kf, out);
}